// SageLayer_87393994539131
// MI455X (gfx1250) — compile-verified
//
#include <hip/hip_runtime.h>

#define N_NODES 50000
#define N_EDGES 800000
#define HIDDEN  128
#define BN_EPS  1e-5f
#define AS_STRIDE 260   // 16-row A staging stride (floats); conflict-free for b64 LDS loads

typedef __attribute__((ext_vector_type(2))) float v2f;
typedef __attribute__((ext_vector_type(8))) float v8f;

__device__ __forceinline__ void atomic_add_f32(float* p, float v) {
    __hip_atomic_fetch_add(p, v, __ATOMIC_RELAXED, __HIP_MEMORY_SCOPE_AGENT);
}

// ---------------- K0: zero workspace ----------------
__global__ void k_zero(float* __restrict__ p, int n) {
    int i = blockIdx.x * blockDim.x + threadIdx.x;
    if (i < n) p[i] = 0.0f;
}

// ---------------- K1: edge scatter (mean-aggregation numerator + degree) ----
// one wave (32 lanes) per edge; lane handles 4 consecutive channels
__global__ void __launch_bounds__(256) k_scatter(
    const float* __restrict__ h, const int* __restrict__ edge_index,
    float* __restrict__ msg_sum, float* __restrict__ deg) {
    int wave = (blockIdx.x * blockDim.x + threadIdx.x) >> 5;
    int lane = threadIdx.x & 31;
    if (wave >= N_EDGES) return;
    int src = edge_index[wave];            // row 0: sources
    int dst = edge_index[N_EDGES + wave];  // row 1: targets
    float4 v = ((const float4*)(h + (size_t)src * HIDDEN))[lane];
    float* ms = msg_sum + (size_t)dst * HIDDEN + lane * 4;
    atomic_add_f32(ms + 0, v.x);
    atomic_add_f32(ms + 1, v.y);
    atomic_add_f32(ms + 2, v.z);
    atomic_add_f32(ms + 3, v.w);
    if (lane == 0) atomic_add_f32(deg + dst, 1.0f);
}

// ---------------- K2: fused dual GEMM via f32 WMMA + BN partial stats -------
// block = 256 threads (8 wave32); block -> 16 nodes, wave w -> cols [16w,16w+16)
// A = [agg | h] (16 x 256) staged in LDS; B = [W_l^T ; W_r^T] read from L2
__global__ void __launch_bounds__(256) k_gemm_wmma(
    const float* __restrict__ msg_sum, const float* __restrict__ deg,
    const float* __restrict__ h,
    const float* __restrict__ W_l, const float* __restrict__ b_l,
    const float* __restrict__ W_r,
    float* __restrict__ lin, float* __restrict__ colsum,
    float* __restrict__ colsumsq) {
    __shared__ float As[16 * AS_STRIDE];

    const int t  = threadIdx.x;
    const int i0 = blockIdx.x * 16;

    // ---- stage A: thread t loads column t for all 16 rows ----
    {
        const int col = t;  // 0..255 ; <128 => agg channel, >=128 => h channel
        for (int row = 0; row < 16; ++row) {
            float v;
            if (col < HIDDEN) {
                float rd = 1.0f / fmaxf(deg[i0 + row], 1.0f);
                v = msg_sum[(size_t)(i0 + row) * HIDDEN + col] * rd;
            } else {
                v = h[(size_t)(i0 + row) * HIDDEN + (col - HIDDEN)];
            }
            As[row * AS_STRIDE + col] = v;
        }
    }
    __syncthreads();

    const int wavei = t >> 5;
    const int lane  = t & 31;
    const int n0    = wavei * 16;
    const int mn    = lane & 15;         // A row / B col within tile
    const int kb    = (lane >> 4) * 2;   // K sub-offset (lanes 16-31 -> K+2,K+3)

    v8f c = {};
    const float* wl   = W_l + (size_t)(n0 + mn) * HIDDEN + kb;
    const float* wr   = W_r + (size_t)(n0 + mn) * HIDDEN + kb;
    const float* arow = As + mn * AS_STRIDE + kb;

    // agg @ W_l^T
#pragma unroll
    for (int kk = 0; kk < HIDDEN; kk += 4) {
        v2f a = *(const v2f*)(arow + kk);   // LDS b64 load, conflict-free
        v2f b = *(const v2f*)(wl + kk);     // L2-resident weights
        c = __builtin_amdgcn_wmma_f32_16x16x4_f32(false, a, false, b,
                                                  (short)0, c, false, false);
    }
    // + h @ W_r^T
#pragma unroll
    for (int kk = 0; kk < HIDDEN; kk += 4) {
        v2f a = *(const v2f*)(arow + HIDDEN + kk);
        v2f b = *(const v2f*)(wr + kk);
        c = __builtin_amdgcn_wmma_f32_16x16x4_f32(false, a, false, b,
                                                  (short)0, c, false, false);
    }

    // ---- epilogue: bias, store lin, accumulate BN column sums ----
    const float bias  = b_l[n0 + mn];
    const int   mbase = (lane >> 4) * 8;   // C layout: VGPR v -> row v (+8 upper half)
    float s = 0.0f, s2 = 0.0f;
#pragma unroll
    for (int v = 0; v < 8; ++v) {
        float val = c[v] + bias;
        lin[(size_t)(i0 + mbase + v) * HIDDEN + n0 + mn] = val;
        s  += val;
        s2 += val * val;
    }
    // lanes l and l^16 hold the same output column (rows 0-7 vs 8-15)
    s  += __shfl_xor(s, 16, 32);
    s2 += __shfl_xor(s2, 16, 32);
    if (lane < 16) {
        atomic_add_f32(&colsum[n0 + mn], s);
        atomic_add_f32(&colsumsq[n0 + mn], s2);
    }
}

// ---------------- K3: batchnorm + relu + residual ---------------------------
__global__ void __launch_bounds__(256) k_bn_relu_res(
    const float* __restrict__ lin, const float* __restrict__ h,
    const float* __restrict__ gamma, const float* __restrict__ beta,
    const float* __restrict__ colsum, const float* __restrict__ colsumsq,
    float* __restrict__ out) {
    int i = blockIdx.x * blockDim.x + threadIdx.x;  // float4 index
    const int total4 = N_NODES * HIDDEN / 4;
    if (i >= total4) return;
    const int c4 = (i * 4) & (HIDDEN - 1);
    float4 l  = ((const float4*)lin)[i];
    float4 hv = ((const float4*)h)[i];
    float li[4] = {l.x, l.y, l.z, l.w};
    float hh[4] = {hv.x, hv.y, hv.z, hv.w};
    float o[4];
    const float invN = 1.0f / (float)N_NODES;
#pragma unroll
    for (int j = 0; j < 4; ++j) {
        int n = c4 + j;
        float mean = colsum[n] * invN;
        float var  = colsumsq[n] * invN - mean * mean;   // biased variance
        float val  = gamma[n] * (li[j] - mean) * rsqrtf(var + BN_EPS) + beta[n];
        val = fmaxf(val, 0.0f);                           // relu
        o[j] = val + hh[j];                               // residual
    }
    ((float4*)out)[i] = make_float4(o[0], o[1], o[2], o[3]);
}

extern "C" void kernel_launch(void* const* d_in, const int* in_sizes, int n_in,
                              void* d_out, int out_size, void* d_ws, size_t ws_size,
                              hipStream_t stream) {
    const float* h    = (const float*)d_in[0];
    const int*   ei   = (const int*)d_in[1];
    const float* W_l  = (const float*)d_in[2];
    const float* b_l  = (const float*)d_in[3];
    const float* W_r  = (const float*)d_in[4];
    const float* gam  = (const float*)d_in[5];
    const float* bet  = (const float*)d_in[6];
    float* out = (float*)d_out;

    // workspace layout: [msg_sum | deg | colsum | colsumsq | lin]
    float* w        = (float*)d_ws;
    float* msg_sum  = w;                                  // N*H
    float* deg      = msg_sum + (size_t)N_NODES * HIDDEN; // N
    float* colsum   = deg + N_NODES;                      // 128
    float* colsumsq = colsum + HIDDEN;                    // 128
    float* lin      = colsumsq + HIDDEN;                  // N*H

    const int n_zero = N_NODES * HIDDEN + N_NODES + 2 * HIDDEN;
    k_zero<<<(n_zero + 255) / 256, 256, 0, stream>>>(w, n_zero);

    // 1 wave per edge, 8 waves per block
    k_scatter<<<(N_EDGES + 7) / 8, 256, 0, stream>>>(h, ei, msg_sum, deg);

    // 16 nodes per block (50000/16 = 3125 exact)
    k_gemm_wmma<<<N_NODES / 16, 256, 0, stream>>>(msg_sum, deg, h, W_l, b_l, W_r,
                                                  lin, colsum, colsumsq);

    const int total4 = N_NODES * HIDDEN / 4;
    k_bn_relu_res<<<(total4 + 255) / 256, 256, 0, stream>>>(lin, h, gam, bet,
                                                            colsum, colsumsq, out);
}